// Model_86586540687779
// MI455X (gfx1250) — compile-verified
//
#include <hip/hip_runtime.h>
#include <hip/hip_bf16.h>
#include <stdint.h>

#define EXPERTS 64
#define SLOTS_PER_BLOCK 256
#define COPY_THREADS 256
#define MAX_H 2048

// CDNA5 async-LDS builtins want pointers to 16B int vectors, with explicit
// address spaces: global = AS(1) (printed as __device__ by HIP clang),
// LDS = AS(3).
typedef int v4i __attribute__((ext_vector_type(4)));
typedef __attribute__((address_space(1))) v4i v4i_gm;
typedef __attribute__((address_space(3))) v4i v4i_lds;

#if defined(__gfx1250__) && \
    __has_builtin(__builtin_amdgcn_global_load_async_to_lds_b128) && \
    __has_builtin(__builtin_amdgcn_global_store_async_from_lds_b128)
#define USE_ASYNC_LDS_COPY 1
#else
#define USE_ASYNC_LDS_COPY 0
#endif

__device__ __forceinline__ void wait_asynccnt0() {
#if __has_builtin(__builtin_amdgcn_s_wait_asynccnt)
    __builtin_amdgcn_s_wait_asynccnt(0);
#else
    asm volatile("s_wait_asynccnt 0" ::: "memory");
#endif
}

// ---------------------------------------------------------------------------
// Kernel A: per-block expert histogram. hist layout: [numBlocks][EXPERTS].
// ---------------------------------------------------------------------------
__global__ void moe_hist_kernel(const int* __restrict__ eidx, int n,
                                int* __restrict__ hist) {
    __shared__ int cnt[EXPERTS];
    const int t = threadIdx.x;
    const int b = blockIdx.x;
    if (t < EXPERTS) cnt[t] = 0;
    __syncthreads();
    const int i = b * SLOTS_PER_BLOCK + t;
    if (i < n) {
        int e = eidx[i];
        atomicAdd(&cnt[e], 1);   // LDS atomic; count is order-independent
    }
    __syncthreads();
    if (t < EXPERTS) hist[b * EXPERTS + t] = cnt[t];
}

// ---------------------------------------------------------------------------
// Kernel B (1 block, EXPERTS threads): convert per-block histograms into
// absolute scatter bases: hist[b][e] = base[e] + sum_{b'<b} hist[b'][e].
// Also emits expert_tokens_count.
// ---------------------------------------------------------------------------
__global__ void moe_scan_kernel(int* __restrict__ hist, int numBlocks,
                                int* __restrict__ out_cnt) {
    __shared__ int total[EXPERTS];
    __shared__ int base[EXPERTS];
    const int e = threadIdx.x;        // one thread per expert
    int run = 0;
    for (int b = 0; b < numBlocks; ++b) {          // coalesced across experts
        int c = hist[b * EXPERTS + e];
        hist[b * EXPERTS + e] = run;
        run += c;
    }
    total[e] = run;
    __syncthreads();
    if (e == 0) {
        int acc = 0;
        for (int k = 0; k < EXPERTS; ++k) { base[k] = acc; acc += total[k]; }
    }
    __syncthreads();
    const int be = base[e];
    for (int b = 0; b < numBlocks; ++b) hist[b * EXPERTS + e] += be;
    out_cnt[e] = total[e];
}

// ---------------------------------------------------------------------------
// Kernel C: stable destination for each slot; writes expanded_row_idx,
// expanded_scale, and the dest->src_row map used by the bulk copy.
// Stable rank = #(j < t in block with same expert), via an LDS sweep.
// ---------------------------------------------------------------------------
__global__ void moe_scatter_kernel(const int* __restrict__ eidx,
                                   const float* __restrict__ scale,
                                   const int* __restrict__ hist,
                                   int n, int K,
                                   int* __restrict__ out_row_idx,
                                   float* __restrict__ out_scale,
                                   int* __restrict__ srcrow) {
    __shared__ int se[SLOTS_PER_BLOCK];
    const int t = threadIdx.x;
    const int b = blockIdx.x;
    const int i = b * SLOTS_PER_BLOCK + t;
    int e = -1;
    if (i < n) e = eidx[i];
    se[t] = e;
    __syncthreads();
    if (i >= n) return;
    int rank = 0;
    for (int j = 0; j < t; ++j) rank += (se[j] == e) ? 1 : 0;
    const int dest = hist[b * EXPERTS + e] + rank;
    out_row_idx[i] = dest;                 // inverse permutation (int bits)
    const int src = i / K;
    srcrow[dest] = src;
    out_scale[dest] = scale[src];
}

// ---------------------------------------------------------------------------
// Kernel D: bulk gather copy. One block per expanded row (8 KB of f32).
// CDNA5 path: async b128 global->LDS, s_wait_asynccnt 0, async b128
// LDS->global. Each lane's stores only read LDS bytes its own loads wrote,
// so the per-wave asynccnt wait is sufficient (no workgroup barrier).
// ---------------------------------------------------------------------------
__global__ void moe_gather_copy_kernel(const float* __restrict__ x,
                                       const int* __restrict__ srcrow,
                                       float* __restrict__ out_x, int H) {
    const int r = blockIdx.x;
    const int src = srcrow[r];
    const float* __restrict__ sp = x + (size_t)src * H;
    float* __restrict__ dp = out_x + (size_t)r * H;
    const int t = threadIdx.x;

#if USE_ASYNC_LDS_COPY
    __shared__ __align__(16) float buf[MAX_H];
    if (H <= MAX_H && (H % (COPY_THREADS * 4)) == 0) {
        for (int off = t * 4; off < H; off += COPY_THREADS * 4) {
            __builtin_amdgcn_global_load_async_to_lds_b128(
                (v4i_gm*)(sp + off), (v4i_lds*)(buf + off), 0, 0);
        }
        wait_asynccnt0();
        for (int off = t * 4; off < H; off += COPY_THREADS * 4) {
            __builtin_amdgcn_global_store_async_from_lds_b128(
                (v4i_gm*)(dp + off), (v4i_lds*)(buf + off), 0, 0);
        }
        wait_asynccnt0();
        return;
    }
#endif
    // Fallback: straight b128 VGPR copy (fully coalesced).
    const float4* __restrict__ s4 = (const float4*)sp;
    float4* __restrict__ d4 = (float4*)dp;
    for (int off = t; off < H / 4; off += COPY_THREADS) d4[off] = s4[off];
}

// ---------------------------------------------------------------------------
extern "C" void kernel_launch(void* const* d_in, const int* in_sizes, int n_in,
                              void* d_out, int out_size, void* d_ws, size_t ws_size,
                              hipStream_t stream) {
    const float* x     = (const float*)d_in[0];
    const int*   eidx  = (const int*)d_in[1];
    const float* scale = (const float*)d_in[2];

    const int num_rows = in_sizes[2];              // scale length
    const int n        = in_sizes[1];              // num_rows * K
    const int K        = n / num_rows;
    const int H        = in_sizes[0] / num_rows;

    const int numBlocks = (n + SLOTS_PER_BLOCK - 1) / SLOTS_PER_BLOCK;

    // Output layout (flat, in reference return order; ints bit-stored):
    float* out_x       = (float*)d_out;                       // [n, H] f32
    int*   out_row_idx = (int*)(out_x + (size_t)n * H);       // [n] i32
    int*   out_cnt     = out_row_idx + n;                     // [EXPERTS] i32
    float* out_scale   = (float*)(out_cnt + EXPERTS);         // [n] f32

    // Workspace layout:
    int* hist   = (int*)d_ws;                                 // numBlocks*EXPERTS
    int* srcrow = hist + (size_t)numBlocks * EXPERTS;         // n ints

    moe_hist_kernel<<<numBlocks, SLOTS_PER_BLOCK, 0, stream>>>(eidx, n, hist);
    moe_scan_kernel<<<1, EXPERTS, 0, stream>>>(hist, numBlocks, out_cnt);
    moe_scatter_kernel<<<numBlocks, SLOTS_PER_BLOCK, 0, stream>>>(
        eidx, scale, hist, n, K, out_row_idx, out_scale, srcrow);
    moe_gather_copy_kernel<<<n, COPY_THREADS, 0, stream>>>(x, srcrow, out_x, H);
}